// STVisualTokenSelection_6150393168247
// MI455X (gfx1250) — compile-verified
//
#include <hip/hip_runtime.h>
#include <stdint.h>

// ---------------------------------------------------------------------------
// STVisualTokenSelection for MI455X (gfx1250, wave32).
//
// softmax over a singleton axis == 1.0 -> the whole LN+MLP branch is dead.
// Output = [cls_feat, ind @ spatial_feat]; ind depends only on the ordering
// of jax.random.normal(key(42)) noise, and every transform after threefry
// (>>9, [1,2) bitcast, affine, erfinv, *sigma, +score) is strictly monotone,
// so we rank raw (threefry_bits >> 9) integers directly.
// ---------------------------------------------------------------------------

#define BF        96      // B * MAX_FRAMES
#define NTOK      197
#define NL        196     // spatial tokens
#define DDIM      512
#define NS        500
#define KSEL      49
#define MPAD      64      // indicator rows padded to 4x16 WMMA M-tiles
#define NHALF     4704000u  // (BF*NS*NL)/2 threefry counter split

#define CK        28      // K chunk (196 = 7 * 28, 28 % 4 == 0)
#define NCHUNK    7
#define BN        128     // N columns per block
#define LBS       144     // padded LDS row stride for B (bank-group shift)

// ---- CDNA5 async global->LDS path (guarded; sync fallback otherwise) ------
#if defined(__has_builtin)
#if __has_builtin(__builtin_amdgcn_global_load_async_to_lds_b32)
#define HAVE_ASYNC_LDS 1
#endif
#endif
#ifndef HAVE_ASYNC_LDS
#define HAVE_ASYNC_LDS 0
#endif

#if HAVE_ASYNC_LDS
typedef __attribute__((address_space(1))) int gas_i32;  // global AS
typedef __attribute__((address_space(3))) int las_i32;  // LDS AS
__device__ __forceinline__ void async_copy_b32(const float* g, float* l) {
  __builtin_amdgcn_global_load_async_to_lds_b32((gas_i32*)g, (las_i32*)l, 0, 0);
}
#if __has_builtin(__builtin_amdgcn_s_wait_asynccnt)
#define WAIT_ASYNCCNT(n) __builtin_amdgcn_s_wait_asynccnt(n)
#else
#define WAIT_ASYNCCNT(n) asm volatile("s_wait_asynccnt %0" ::"i"(n))
#endif
#endif

__device__ __forceinline__ uint32_t rotl32(uint32_t x, int r) {
  return (x << r) | (x >> (32 - r));
}

// threefry2x32-20 with key = jax.random.key(42) -> (0, 42)
__device__ __forceinline__ void threefry2x32_42(uint32_t x0, uint32_t x1,
                                                uint32_t& o0, uint32_t& o1) {
  const uint32_t k0 = 0u, k1 = 42u;
  const uint32_t k2 = k0 ^ k1 ^ 0x1BD11BDAu;
  const uint32_t ks[3] = {k0, k1, k2};
  const int rotA[4] = {13, 15, 26, 6};
  const int rotB[4] = {17, 29, 16, 24};
  x0 += k0; x1 += k1;
#pragma unroll
  for (int i = 0; i < 5; ++i) {
#pragma unroll
    for (int j = 0; j < 4; ++j) {
      int r = (i & 1) ? rotB[j] : rotA[j];
      x0 += x1; x1 = rotl32(x1, r); x1 ^= x0;
    }
    x0 += ks[(i + 1) % 3];
    x1 += ks[(i + 2) % 3] + (uint32_t)(i + 1);
  }
  o0 = x0; o1 = x1;
}

// JAX counter-mode split: element i<h takes out0 of pair (i, i+h), else out1.
__device__ __forceinline__ uint32_t jax_bits(uint32_t fi) {
  uint32_t o0, o1;
  if (fi < NHALF) { threefry2x32_42(fi, fi + NHALF, o0, o1); return o0; }
  threefry2x32_42(fi - NHALF, fi, o0, o1); return o1;
}

// ---------------------------------------------------------------------------
// Kernel 1: per bf row, histogram sorted top-49 noise indices over 500
// samples. One workgroup per bf; each wave owns one sample at a time
// (wave-synchronous shfl broadcast + ballot prefix-popcount). Indicator
// matrix written zero-padded to 64 rows for unconditional WMMA A loads.
// ---------------------------------------------------------------------------
__global__ __launch_bounds__(256) void topk_hist_kernel(float* __restrict__ ind) {
  __shared__ unsigned int cnt[KSEL * NL];  // 9604 u32 = 38.4 KB LDS
  const int tid = threadIdx.x;
  const int bf  = blockIdx.x;

  for (int i = tid; i < KSEL * NL; i += 256) cnt[i] = 0u;
  __syncthreads();

  const int lane = tid & 31;
  const int wave = tid >> 5;
  const unsigned int ltmask = (lane == 0) ? 0u : (0xFFFFFFFFu >> (32 - lane));

  for (int s = wave; s < NS; s += 8) {
    uint32_t myk[7];  // lane holds indices i = r*32 + lane
#pragma unroll
    for (int r = 0; r < 7; ++r) {
      const int i = r * 32 + lane;
      uint32_t key = 0u;
      if (i < NL) key = jax_bits((uint32_t)((bf * NS + s) * NL + i)) >> 9;
      myk[r] = key;
    }

    // rank(i) = #{j : key_j > key_i} + #{j < i : key_j == key_i}
    int rank[7] = {0, 0, 0, 0, 0, 0, 0};
    for (int j = 0; j < NL; ++j) {
      const uint32_t kj =
          (uint32_t)__shfl((unsigned int)myk[j >> 5], j & 31, 32);
#pragma unroll
      for (int r = 0; r < 7; ++r) {
        const int i = r * 32 + lane;
        rank[r] += ((kj > myk[r]) | ((kj == myk[r]) & (j < i))) ? 1 : 0;
      }
    }

    // exactly KSEL ranks < KSEL (total order); slot = prefix count below i.
    unsigned int m[7];
#pragma unroll
    for (int r = 0; r < 7; ++r) {
      const int i = r * 32 + lane;
      const bool sel = (i < NL) && (rank[r] < KSEL);
      m[r] = (unsigned int)(__ballot(sel) & 0xFFFFFFFFull);
    }
    int pre = 0;
#pragma unroll
    for (int r = 0; r < 7; ++r) {
      const int i = r * 32 + lane;
      const bool sel = (i < NL) && (rank[r] < KSEL);
      const int slot = pre + __popc(m[r] & ltmask);
      if (sel) atomicAdd(&cnt[slot * NL + i], 1u);
      pre += __popc(m[r]);
    }
  }
  __syncthreads();

  const float inv = 1.0f / (float)NS;
  float* dst = ind + (size_t)bf * (MPAD * NL);
  for (int i = tid; i < MPAD * NL; i += 256)
    dst[i] = (i < KSEL * NL) ? (float)cnt[i] * inv : 0.0f;
}

// ---------------------------------------------------------------------------
// Kernel 2: out[bf,0,:] = cls; out[bf,1+k,:] = ind[bf] @ spatial_feat[bf].
// 96-batch 64x196 @ 196x512 f32 GEMM via V_WMMA_F32_16X16X4_F32.
// Grid (96,4): 128 columns per block; double-buffered async global->LDS
// staging of A (64x28) and B (28x128) chunks, pipelined against WMMA.
// ---------------------------------------------------------------------------
typedef float v2f __attribute__((ext_vector_type(2)));
typedef float v8f __attribute__((ext_vector_type(8)));

__global__ __launch_bounds__(256) void select_gemm_kernel(
    const float* __restrict__ x, const float* __restrict__ ind,
    float* __restrict__ out) {
  __shared__ float lA[2][MPAD * CK];  // 2 * 7 KB
  __shared__ float lB[2][CK * LBS];   // 2 * 15.75 KB

  const int bf = blockIdx.x;
  const int nbase128 = blockIdx.y * BN;
  const int tid = threadIdx.x;

  // cls token passthrough for this column chunk
  if (tid < BN) {
    const int col = nbase128 + tid;
    out[(size_t)(bf * 50) * DDIM + col] = x[(size_t)(bf * NTOK) * DDIM + col];
  }

  const int lane = tid & 31;
  const int wave = tid >> 5;
  const int lo = lane & 15;
  const int hi = lane >> 4;       // 0/1: K-pair select (A/B), M-half (C/D)
  const int mrow = (wave & 3) * 16 + lo;
  const int ngrp = wave >> 2;     // 0/1 -> 4 N-tiles each

  const float* Ag = ind + (size_t)bf * (MPAD * NL);
  const float* Bg = x + ((size_t)bf * NTOK + 1) * DDIM + nbase128;

  v8f acc[4] = {v8f{}, v8f{}, v8f{}, v8f{}};

  // stage chunk c into buffer bufsel: 21 transfers per thread (7 A + 14 B)
  auto stage = [&](int c, int bufsel) {
    const int kc = c * CK;
    float* la = &lA[bufsel][0];
    float* lb = &lB[bufsel][0];
#pragma unroll
    for (int q = 0; q < 7; ++q) {              // A: 64 x 28
      const int idx = q * 256 + tid;
      const int row = idx / CK, col = idx - row * CK;
      const float* g = &Ag[row * NL + kc + col];
#if HAVE_ASYNC_LDS
      async_copy_b32(g, &la[idx]);
#else
      la[idx] = *g;
#endif
    }
#pragma unroll
    for (int q = 0; q < 14; ++q) {             // B: 28 x 128 (stride 144)
      const int idx = q * 256 + tid;
      const int row = idx >> 7, col = idx & (BN - 1);
      const float* g = &Bg[(size_t)(kc + row) * DDIM + col];
#if HAVE_ASYNC_LDS
      async_copy_b32(g, &lb[row * LBS + col]);
#else
      lb[row * LBS + col] = *g;
#endif
    }
  };

  auto compute = [&](int bufsel) {
    const float* la = &lA[bufsel][0];
    const float* lb = &lB[bufsel][0];
#pragma unroll
    for (int k0 = 0; k0 < CK; k0 += 4) {
      const int kk = k0 + 2 * hi;  // A/B: lanes0-15 K={k,k+1}, lanes16-31 +2
      const v2f a = *(const v2f*)&la[mrow * CK + kk];
#pragma unroll
      for (int t = 0; t < 4; ++t) {
        const int nc = (ngrp * 4 + t) * 16 + lo;
        v2f b;
        b.x = lb[kk * LBS + nc];
        b.y = lb[(kk + 1) * LBS + nc];
        acc[t] = __builtin_amdgcn_wmma_f32_16x16x4_f32(
            false, a, false, b, (short)0, acc[t], false, false);
      }
    }
  };

#if HAVE_ASYNC_LDS
  stage(0, 0);
  for (int c = 0; c < NCHUNK; ++c) {
    if (c + 1 < NCHUNK) {
      stage(c + 1, (c + 1) & 1);   // issue ahead; ASYNCcnt is in-order,
      WAIT_ASYNCCNT(21);           // so <=21 outstanding => chunk c landed
    } else {
      WAIT_ASYNCCNT(0);
    }
    __syncthreads();
    compute(c & 1);
    __syncthreads();
  }
#else
  for (int c = 0; c < NCHUNK; ++c) {
    stage(c, 0);
    __syncthreads();
    compute(0);
    __syncthreads();
  }
#endif

  // C/D layout: lanes0-15 vgpr r -> M = r; lanes16-31 -> M = 8 + r; N = lo.
#pragma unroll
  for (int t = 0; t < 4; ++t) {
    const int nb = nbase128 + (ngrp * 4 + t) * 16;
#pragma unroll
    for (int r = 0; r < 8; ++r) {
      const int mm = (mrow & ~15) + hi * 8 + r;
      if (mm < KSEL)
        out[((size_t)bf * 50 + 1 + mm) * DDIM + nb + lo] = acc[t][r];
    }
  }
}

// ---------------------------------------------------------------------------
extern "C" void kernel_launch(void* const* d_in, const int* in_sizes, int n_in,
                              void* d_out, int out_size, void* d_ws,
                              size_t ws_size, hipStream_t stream) {
  (void)in_sizes; (void)n_in; (void)out_size; (void)ws_size;
  const float* x = (const float*)d_in[0];  // (8, 12*197, 512) f32
  // d_in[1..5] (ln_w, ln_b, w_in, w_out1, w_out2) are mathematically dead:
  // softmax over the singleton output axis is identically 1.0.
  float* ind = (float*)d_ws;   // (96, 64, 196) f32, rows 49..63 zero-padded
  float* out = (float*)d_out;  // (8, 600, 512) f32

  topk_hist_kernel<<<dim3(BF), dim3(256), 0, stream>>>(ind);
  select_gemm_kernel<<<dim3(BF, 4), dim3(256), 0, stream>>>(x, ind, out);
}